// LogisticGaussianHMM_76862734730021
// MI455X (gfx1250) — compile-verified
//
#include <hip/hip_runtime.h>
#include <math.h>

// ---------------------------------------------------------------------------
// LogisticGaussianHMM forward-backward on MI455X (gfx1250, wave32, WMMA)
//   T=512 B=32 S=64 A=8 D=32
// Design:
//  - emissions: fused double fp32-WMMA GEMM  [16384x32]x[32x64]
//  - transitions: rebuilt per 16-step window with WMMA (M=16 steps, K=8, N=4096)
//    into a bank-rotated 272KB LDS window (stride 68 floats per 64-wide row)
//  - scans: one workgroup per batch element; per-step reductions use 4x16
//    partials + wave32 shuffle butterflies (latency-critical path)
// ---------------------------------------------------------------------------

typedef __attribute__((ext_vector_type(2))) float v2f;
typedef __attribute__((ext_vector_type(8))) float v8f;

#define T_DIM   512
#define B_DIM   32
#define S_DIM   64
#define A_DIM   8
#define D_DIM   32
#define N_ROWS  (T_DIM * B_DIM)     // 16384
#define NSTEPS  (T_DIM - 1)         // 511 transition steps
#define WIN     16                  // timesteps per WMMA transition window
#define NWIN    ((NSTEPS + WIN - 1) / WIN)  // 32
#define SS      (S_DIM * S_DIM)     // 4096
#define RSTR    68                  // LDS row stride (floats): 272B -> 16B aligned,
                                    // rotates banks by 4 per row (<=2-way conflicts)
#define LDS_FLOATS (WIN * S_DIM * RSTR)     // 69632 floats = 272 KB
#define EPS_F   1e-6f
#define LOG2PI  1.8378770664093453f

// V_WMMA_F32_16X16X4_F32: D(16x16 f32, 8 VGPR) = A(16x4, 2 VGPR) x B(4x16, 2 VGPR) + C
__device__ __forceinline__ v8f wmma4(v2f a, v2f b, v8f c) {
  return __builtin_amdgcn_wmma_f32_16x16x4_f32(
      /*neg_a=*/false, a, /*neg_b=*/false, b,
      /*c_mod=*/(short)0, c, /*reuse_a=*/false, /*reuse_b=*/false);
}

// ---------------------------------------------------------------------------
// Kernel P: per-state tables
// ---------------------------------------------------------------------------
__global__ void prep_kernel(const float* __restrict__ mu,
                            const float* __restrict__ logv,
                            const float* __restrict__ init_logits,
                            float* __restrict__ invvar, float* __restrict__ w1,
                            float* __restrict__ cconst, float* __restrict__ logp0) {
  int tid = threadIdx.x;  // 256 threads
  for (int i = tid; i < S_DIM * D_DIM; i += 256) {
    float iv  = __expf(-logv[i]);
    invvar[i] = iv;
    w1[i]     = mu[i] * iv;
  }
  __syncthreads();
  if (tid < S_DIM) {
    float c = 0.f;
    for (int d = 0; d < D_DIM; ++d) {
      int idx = tid * D_DIM + d;
      c += mu[idx] * mu[idx] * invvar[idx] + logv[idx] + LOG2PI;
    }
    cconst[tid] = -0.5f * c;
    float mx = -1e30f;
    for (int s = 0; s < S_DIM; ++s) mx = fmaxf(mx, init_logits[s]);
    float sm = 0.f;
    for (int s = 0; s < S_DIM; ++s) sm += __expf(init_logits[s] - mx);
    logp0[tid] = init_logits[tid] - mx - __logf(sm);
  }
}

// ---------------------------------------------------------------------------
// Kernel E: emission log-likelihoods as a fused double WMMA GEMM
//   logp_x[n,s] = X[n,:]·w1[s,:] + (-0.5 X^2)[n,:]·invvar[s,:] + cconst[s]
// ---------------------------------------------------------------------------
__global__ void emis_kernel(const float* __restrict__ x,
                            const float* __restrict__ invvar,
                            const float* __restrict__ w1,
                            const float* __restrict__ cconst,
                            float* __restrict__ logp_x) {
  int tileRow = blockIdx.x;       // 0..N_ROWS/16-1
  int tileCol = blockIdx.y;       // 0..S/16-1
  int lane = threadIdx.x;         // 32 (one wave)
  int m    = lane & 15;
  int half = lane >> 4;
  int row  = tileRow * 16 + m;
  int scol = tileCol * 16 + (lane & 15);
  const float* xr  = x + row * D_DIM;
  const float* w1c = w1 + scol * D_DIM;
  const float* ivc = invvar + scol * D_DIM;

  v8f acc = {};
#pragma unroll
  for (int k0 = 0; k0 < D_DIM; k0 += 4) {
    int ka = k0 + half * 2;  // A 16x4 f32 layout: lane m holds K = half*2, half*2+1
    float x0 = xr[ka], x1 = xr[ka + 1];
    v2f afrag  = {x0, x1};
    v2f afrag2 = {-0.5f * x0 * x0, -0.5f * x1 * x1};
    v2f bfrag1 = {w1c[ka], w1c[ka + 1]};   // B[k][s] = w1[s][k]
    v2f bfrag2 = {ivc[ka], ivc[ka + 1]};
    acc = wmma4(afrag,  bfrag1, acc);
    acc = wmma4(afrag2, bfrag2, acc);
  }
  float cc = cconst[scol];
#pragma unroll
  for (int v = 0; v < 8; ++v) {            // C: VGPR v -> M = v + 8*half, N = lane%16
    int r = tileRow * 16 + v + 8 * half;
    logp_x[r * S_DIM + scol] = acc[v] + cc;
  }
}

// ---------------------------------------------------------------------------
// Window transition builder (shared by fwd/bwd):
//   logits[step][i][j] = w_base[i,j] + sum_a a[t0+step,b,a]*w_act[a,i,j]
//   WMMA: M=16 steps, K=8 (two K=4 chunks), N=4096 in 16-wide tiles.
//   LDS row (step,i) stored at (step*64+i)*RSTR for bank rotation.
// ---------------------------------------------------------------------------
__device__ __forceinline__ void build_window_logits(
    float* __restrict__ tl, const float (*s_avec)[A_DIM],
    const float* __restrict__ w_base, const float* __restrict__ w_act,
    int tid) {
  int lane = tid & 31;
  int wv   = tid >> 5;     // wave 0..7
  int half = lane >> 4;
  int mrow = lane & 15;    // A-matrix row = timestep in window
  v2f afr0 = {s_avec[mrow][half * 2],     s_avec[mrow][half * 2 + 1]};      // K 0..3
  v2f afr1 = {s_avec[mrow][4 + half * 2], s_avec[mrow][4 + half * 2 + 1]};  // K 4..7
  for (int nt = 0; nt < 32; ++nt) {
    int ncol = (wv * 32 + nt) * 16 + (lane & 15);
    v2f b0 = {w_act[(half * 2) * SS + ncol],     w_act[(half * 2 + 1) * SS + ncol]};
    v2f b1 = {w_act[(4 + half * 2) * SS + ncol], w_act[(4 + half * 2 + 1) * SS + ncol]};
    v8f acc = {};
    acc = wmma4(afr0, b0, acc);
    acc = wmma4(afr1, b1, acc);
    float wb = w_base[ncol];
    int irow = ncol >> 6, jcol = ncol & 63;
#pragma unroll
    for (int v = 0; v < 8; ++v) {
      int st = v + 8 * half;               // C row = timestep
      tl[(st * S_DIM + irow) * RSTR + jcol] = acc[v] + wb;
    }
  }
}

// ---------------------------------------------------------------------------
// Kernel F: forward scan. One workgroup (8 waves) per batch element.
//   alpha[t+1] = softmax(logp_x[t+1] + log(alpha[t]·trans[t] + eps))
// ---------------------------------------------------------------------------
__global__ void fwd_kernel(const float* __restrict__ a,
                           const float* __restrict__ w_base,
                           const float* __restrict__ w_act,
                           const float* __restrict__ logp_x,
                           const float* __restrict__ logp0,
                           float* __restrict__ alpha) {
  extern __shared__ float tl[];            // LDS_FLOATS (272 KB)
  __shared__ float s_alpha[S_DIM];
  __shared__ float s_tmp[S_DIM];
  __shared__ float s_part[256];
  __shared__ float s_redA[8];
  __shared__ float s_redB[8];
  __shared__ float s_avec[WIN][A_DIM];
  int b   = blockIdx.x;
  int tid = threadIdx.x;                   // 256 threads = 8 wave32

  // alpha_0 = softmax(logp_x[0] + logp0)   (one-time init)
  float v0 = 0.f;
  if (tid < S_DIM) {
    v0 = logp_x[(0 * B_DIM + b) * S_DIM + tid] + logp0[tid];
    s_tmp[tid] = v0;
  }
  __syncthreads();
  if (tid < S_DIM) {
    float mx = -1e30f;
    for (int k = 0; k < S_DIM; ++k) mx = fmaxf(mx, s_tmp[k]);
    float sm = 0.f;
    for (int k = 0; k < S_DIM; ++k) sm += __expf(s_tmp[k] - mx);
    float av = __expf(v0 - mx) / sm;
    s_alpha[tid] = av;
    alpha[(0 * B_DIM + b) * S_DIM + tid] = av;
  }
  __syncthreads();

  for (int w = 0; w < NWIN; ++w) {
    int t0 = w * WIN;
    int nsteps = min(WIN, NSTEPS - t0);
    if (tid < WIN * A_DIM) {               // stage action window
      int st = tid / A_DIM, k = tid % A_DIM;
      int t = t0 + st;
      s_avec[st][k] = (t < NSTEPS) ? a[(t * B_DIM + b) * A_DIM + k] : 0.f;
    }
    if (tid == 0 && w + 1 < NWIN)          // warm L2 for next window's emissions
      __builtin_prefetch(&logp_x[((t0 + WIN + 1) * B_DIM + b) * S_DIM], 0, 1);
    __syncthreads();

    build_window_logits(tl, s_avec, w_base, w_act, tid);
    __syncthreads();

    // softmax each (step,i) row over j -- float4 LDS (b128), 3 passes
    for (int r = tid; r < WIN * S_DIM; r += 256) {
      float4* rowp = (float4*)(tl + r * RSTR);
      float mx = -1e30f;
#pragma unroll
      for (int j = 0; j < 16; ++j) {
        float4 v4 = rowp[j];
        mx = fmaxf(mx, fmaxf(fmaxf(v4.x, v4.y), fmaxf(v4.z, v4.w)));
      }
      float sm = 0.f;
#pragma unroll
      for (int j = 0; j < 16; ++j) {
        float4 v4 = rowp[j];
        sm += __expf(v4.x - mx) + __expf(v4.y - mx) +
              __expf(v4.z - mx) + __expf(v4.w - mx);
      }
      float inv = 1.f / sm;
#pragma unroll
      for (int j = 0; j < 16; ++j) {
        float4 v4 = rowp[j];
        v4.x = __expf(v4.x - mx) * inv;
        v4.y = __expf(v4.y - mx) * inv;
        v4.z = __expf(v4.z - mx) * inv;
        v4.w = __expf(v4.w - mx) * inv;
        rowp[j] = v4;
      }
    }
    __syncthreads();

    // sequential recursion: thread = (j, quarter q of i-range)
    int j = tid & 63, q = tid >> 6;
    for (int s = 0; s < nsteps; ++s) {
      int t = t0 + s;
      const float* trb = tl + (s * S_DIM) * RSTR;
      float part = 0.f;
#pragma unroll
      for (int ii = 0; ii < 16; ++ii) {
        int i = q * 16 + ii;
        part += trb[i * RSTR + j] * s_alpha[i];   // lanes vary j -> conflict-free
      }
      s_part[tid] = part;
      __syncthreads();
      float sum = EPS_F + s_part[j] + s_part[64 + j] + s_part[128 + j] + s_part[192 + j];
      float val = logp_x[((t + 1) * B_DIM + b) * S_DIM + j] + __logf(sum);
      // max over 64 distinct j (each wave holds 32 distinct; waves 0/1 cover all)
      float m = val;
#pragma unroll
      for (int off = 16; off >= 1; off >>= 1) m = fmaxf(m, __shfl_xor(m, off, 32));
      if ((tid & 31) == 0) s_redA[tid >> 5] = m;
      __syncthreads();
      float mx = fmaxf(s_redA[0], s_redA[1]);
      float ex = __expf(val - mx);
      float sme = ex;
#pragma unroll
      for (int off = 16; off >= 1; off >>= 1) sme += __shfl_xor(sme, off, 32);
      if ((tid & 31) == 0) s_redB[tid >> 5] = sme;
      __syncthreads();
      float av = ex / (s_redB[0] + s_redB[1]);
      if (q == 0) {
        s_alpha[j] = av;
        alpha[((t + 1) * B_DIM + b) * S_DIM + j] = av;
      }
      __syncthreads();
    }
  }
}

// ---------------------------------------------------------------------------
// Kernel B: backward scan + posterior, one workgroup per batch element.
//   logb[i] = mask * logsumexp_j(lx[t+1][j] + log(trans[t][i][j]+eps) + logb[j])
//   q_z[t]  = softmax(log(alpha[t]+eps) + logb)
// ---------------------------------------------------------------------------
__global__ void bwd_kernel(const float* __restrict__ a,
                           const float* __restrict__ w_base,
                           const float* __restrict__ w_act,
                           const float* __restrict__ mask,
                           const float* __restrict__ logp_x,
                           const float* __restrict__ alpha,
                           float* __restrict__ qz) {
  extern __shared__ float tl[];            // LDS_FLOATS (272 KB)
  __shared__ float s_logb[S_DIM];
  __shared__ float s_vec[S_DIM];
  __shared__ float s_tmp[S_DIM];
  __shared__ float s_part[256];
  __shared__ float s_partB[256];
  __shared__ float s_redA[8];
  __shared__ float s_redB[8];
  __shared__ float s_avec[WIN][A_DIM];
  int b   = blockIdx.x;
  int tid = threadIdx.x;

  // q_z[T-1] = softmax(log(alpha[T-1]+eps)); logb = 0   (one-time init)
  float qv0 = 0.f;
  if (tid < S_DIM) {
    s_logb[tid] = 0.f;
    qv0 = __logf(alpha[((T_DIM - 1) * B_DIM + b) * S_DIM + tid] + EPS_F);
    s_tmp[tid] = qv0;
  }
  __syncthreads();
  if (tid < S_DIM) {
    float mx = -1e30f;
    for (int k = 0; k < S_DIM; ++k) mx = fmaxf(mx, s_tmp[k]);
    float sm = 0.f;
    for (int k = 0; k < S_DIM; ++k) sm += __expf(s_tmp[k] - mx);
    qz[((T_DIM - 1) * B_DIM + b) * S_DIM + tid] = __expf(qv0 - mx) / sm;
  }
  __syncthreads();

  for (int w = NWIN - 1; w >= 0; --w) {
    int t0 = w * WIN;
    int nsteps = min(WIN, NSTEPS - t0);
    if (tid < WIN * A_DIM) {
      int st = tid / A_DIM, k = tid % A_DIM;
      int t = t0 + st;
      s_avec[st][k] = (t < NSTEPS) ? a[(t * B_DIM + b) * A_DIM + k] : 0.f;
    }
    __syncthreads();

    build_window_logits(tl, s_avec, w_base, w_act, tid);
    __syncthreads();

    // rows -> log(softmax + eps), float4 LDS
    for (int r = tid; r < WIN * S_DIM; r += 256) {
      float4* rowp = (float4*)(tl + r * RSTR);
      float mx = -1e30f;
#pragma unroll
      for (int j = 0; j < 16; ++j) {
        float4 v4 = rowp[j];
        mx = fmaxf(mx, fmaxf(fmaxf(v4.x, v4.y), fmaxf(v4.z, v4.w)));
      }
      float sm = 0.f;
#pragma unroll
      for (int j = 0; j < 16; ++j) {
        float4 v4 = rowp[j];
        sm += __expf(v4.x - mx) + __expf(v4.y - mx) +
              __expf(v4.z - mx) + __expf(v4.w - mx);
      }
      float inv = 1.f / sm;
#pragma unroll
      for (int j = 0; j < 16; ++j) {
        float4 v4 = rowp[j];
        v4.x = __logf(__expf(v4.x - mx) * inv + EPS_F);
        v4.y = __logf(__expf(v4.y - mx) * inv + EPS_F);
        v4.z = __logf(__expf(v4.z - mx) * inv + EPS_F);
        v4.w = __logf(__expf(v4.w - mx) * inv + EPS_F);
        rowp[j] = v4;
      }
    }
    __syncthreads();

    // sequential recursion: thread = (i, quarter q of j-range)
    int i = tid & 63, q = tid >> 6;
    for (int s = nsteps - 1; s >= 0; --s) {
      int t = t0 + s;
      if (q == 0)
        s_vec[i] = logp_x[((t + 1) * B_DIM + b) * S_DIM + i] + s_logb[i];
      __syncthreads();
      const float* ltr = tl + (s * S_DIM + i) * RSTR;
      float pm = -1e30f;
#pragma unroll
      for (int jj = 0; jj < 16; ++jj) {
        int j2 = q * 16 + jj;
        pm = fmaxf(pm, s_vec[j2] + ltr[j2]);
      }
      s_part[tid] = pm;
      __syncthreads();
      float m4 = fmaxf(fmaxf(s_part[i], s_part[64 + i]),
                       fmaxf(s_part[128 + i], s_part[192 + i]));
      float ps = 0.f;
#pragma unroll
      for (int jj = 0; jj < 16; ++jj) {
        int j2 = q * 16 + jj;
        ps += __expf(s_vec[j2] + ltr[j2] - m4);
      }
      s_partB[tid] = ps;
      __syncthreads();
      float sm4 = s_partB[i] + s_partB[64 + i] + s_partB[128 + i] + s_partB[192 + i];
      float newlb = m4 + __logf(sm4);
      float mk = mask[(t + 1) * B_DIM + b];
      newlb = (mk == 1.f) ? newlb : 0.f;
      if (q == 0) s_logb[i] = newlb;
      __syncthreads();
      // q_z[t] = softmax(log(alpha[t]+eps) + logb)
      float qv = __logf(alpha[(t * B_DIM + b) * S_DIM + i] + EPS_F) + s_logb[i];
      float m = qv;
#pragma unroll
      for (int off = 16; off >= 1; off >>= 1) m = fmaxf(m, __shfl_xor(m, off, 32));
      if ((tid & 31) == 0) s_redA[tid >> 5] = m;
      __syncthreads();
      float mx = fmaxf(s_redA[0], s_redA[1]);
      float ex = __expf(qv - mx);
      float sme = ex;
#pragma unroll
      for (int off = 16; off >= 1; off >>= 1) sme += __shfl_xor(sme, off, 32);
      if ((tid & 31) == 0) s_redB[tid >> 5] = sme;
      __syncthreads();
      if (q == 0)
        qz[(t * B_DIM + b) * S_DIM + i] = ex / (s_redB[0] + s_redB[1]);
      __syncthreads();
    }
  }
}

// ---------------------------------------------------------------------------
extern "C" void kernel_launch(void* const* d_in, const int* in_sizes, int n_in,
                              void* d_out, int out_size, void* d_ws, size_t ws_size,
                              hipStream_t stream) {
  (void)in_sizes; (void)n_in; (void)out_size; (void)ws_size;
  const float* x           = (const float*)d_in[0];  // [T,B,D]
  const float* a           = (const float*)d_in[1];  // [T,B,A]
  const float* mask        = (const float*)d_in[2];  // [T,B]
  const float* mu          = (const float*)d_in[3];  // [S,D]
  const float* logv        = (const float*)d_in[4];  // [S,D]
  const float* init_logits = (const float*)d_in[5];  // [S]
  const float* w_base      = (const float*)d_in[6];  // [S,S]
  const float* w_act       = (const float*)d_in[7];  // [A,S,S]
  float* qz = (float*)d_out;                         // [T,B,S]

  float* ws      = (float*)d_ws;                     // ~8.4 MB used
  float* invvar  = ws;
  float* w1      = invvar + S_DIM * D_DIM;
  float* cconst  = w1 + S_DIM * D_DIM;
  float* logp0   = cconst + S_DIM;
  float* logp_x  = logp0 + S_DIM;                    // [T*B*S]
  float* alpha   = logp_x + (size_t)N_ROWS * S_DIM;  // [T*B*S]

  prep_kernel<<<1, 256, 0, stream>>>(mu, logv, init_logits, invvar, w1, cconst, logp0);
  emis_kernel<<<dim3(N_ROWS / 16, S_DIM / 16), 32, 0, stream>>>(x, invvar, w1, cconst, logp_x);
  size_t shbytes = (size_t)LDS_FLOATS * sizeof(float);  // 272 KB of the WGP's 320 KB LDS
  fwd_kernel<<<B_DIM, 256, shbytes, stream>>>(a, w_base, w_act, logp_x, logp0, alpha);
  bwd_kernel<<<B_DIM, 256, shbytes, stream>>>(a, w_base, w_act, mask, logp_x, alpha, qz);
}